// GASF_68461778698455
// MI455X (gfx1250) — compile-verified
//
#include <hip/hip_runtime.h>

typedef __attribute__((ext_vector_type(2))) float v2f;
typedef __attribute__((ext_vector_type(8))) float v8f;

#define GASF_L   128
#define GASF_OUT 65
#define GASF_EPS 1e-6f

// One workgroup (256 threads = 8 wave32) per (n,c) row.
// Phase 1: LDS tree reductions for first/last nonzero and masked min/max.
// Phase 2: c/s vectors -> rank-2 outer product G = c c^T - s s^T via
//          V_WMMA_F32_16X16X4_F32; 25 tiles of 16x16 cover the 65x65 output.
//          Store path specialized per tile class so interior tiles issue
//          8 unconditional coalesced b32 stores (no per-row branches).
__global__ __launch_bounds__(256) void gasf_wmma_kernel(const float* __restrict__ x,
                                                        float* __restrict__ out) {
  __shared__ int   rFirst[256];
  __shared__ int   rLast[256];
  __shared__ float rMin[256];
  __shared__ float rMax[256];
  __shared__ float cA[80];   // c padded to 5*16
  __shared__ float sA[80];   // s padded to 5*16

  const int b = blockIdx.x;          // flat (n*C + c)
  const int t = threadIdx.x;
  const float* xr = x + (size_t)b * GASF_L;

  // ---------------- phase 1: stats ----------------
  float xv = 0.0f;
  bool nz = false;
  if (t < GASF_L) {
    xv = xr[t];
    nz = (xv != 0.0f);
  }

  rFirst[t] = nz ? t : GASF_L;   // min-reduce -> first nonzero
  rLast[t]  = nz ? t : -1;       // max-reduce -> last nonzero
  __syncthreads();
  for (int s = 128; s > 0; s >>= 1) {
    if (t < s) {
      rFirst[t] = min(rFirst[t], rFirst[t + s]);
      rLast[t]  = max(rLast[t],  rLast[t + s]);
    }
    __syncthreads();
  }
  const int first = rFirst[0];
  const int last  = rLast[0];

  const bool valid = (t < GASF_L) && (t >= first) && (t <= last);
  rMin[t] = valid ? xv :  INFINITY;
  rMax[t] = valid ? xv : -INFINITY;
  __syncthreads();
  for (int s = 128; s > 0; s >>= 1) {
    if (t < s) {
      rMin[t] = fminf(rMin[t], rMin[t + s]);
      rMax[t] = fmaxf(rMax[t], rMax[t + s]);
    }
    __syncthreads();
  }
  float vmin = rMin[0];
  float vmax = rMax[0];

  vmin = fminf((vmin ==  INFINITY) ? 0.0f : vmin, 0.0f);
  vmax = fmaxf((vmax == -INFINITY) ? 0.0f : vmax, 0.0f);
  const float inv2 = 2.0f / fmaxf(vmax - vmin, GASF_EPS);

  // c_i = clip(x_norm_i), s_i = sqrt(1 - c_i^2)  (invalid lanes: x_norm=0 -> c=0,s=1,
  // matching reference: cos(pi/2 + phi_j) = -sin(phi_j))
  if (t < 80) {
    float cv = 0.0f, sv = 0.0f;
    if (t < GASF_OUT) {
      const float xn = valid ? ((xv - vmin) * inv2 - 1.0f) : 0.0f;
      cv = fminf(fmaxf(xn, -1.0f + GASF_EPS), 1.0f - GASF_EPS);
      sv = sqrtf(fmaxf(1.0f - cv * cv, 0.0f));
    }
    cA[t] = cv;   // pad rows 65..79 are zero; never stored
    sA[t] = sv;
  }
  __syncthreads();

  // ---------------- phase 2: rank-2 GEMM via WMMA ----------------
  const int  wv   = t >> 5;        // wave id 0..7
  const int  lane = t & 31;
  const int  m    = lane & 15;
  const bool hi   = lane >= 16;    // lanes 16..31 carry K=2,3 -> zeros

  float* const op = out + (size_t)b * (GASF_OUT * GASF_OUT);

  for (int tile = wv; tile < 25; tile += 8) {
    const int ti = tile / 5;       // row tile
    const int tj = tile % 5;       // col tile

    // A (16x4, MxK): row M = m, K0 = c_M, K1 = -s_M, K2=K3=0
    // B (4x16, KxN): col N = m, K0 = c_N, K1 =  s_N, K2=K3=0
    v2f a  = {0.0f, 0.0f};
    v2f bb = {0.0f, 0.0f};
    if (!hi) {
      a.x  =  cA[ti * 16 + m];
      a.y  = -sA[ti * 16 + m];
      bb.x =  cA[tj * 16 + m];
      bb.y =  sA[tj * 16 + m];
    }
    v8f acc = {};
    // D = A x B + 0  ->  G[M,N] = c_M c_N - s_M s_N
    acc = __builtin_amdgcn_wmma_f32_16x16x4_f32(false, a, false, bb, (short)0, acc,
                                                false, false);

    // C/D layout: VGPR v -> (M = ti*16 + v + (hi?8:0), N = tj*16 + (lane&15))
    const int col = tj * 16 + m;
    if (tj < 4) {
      if (ti < 4) {
        // interior tile: all 16 rows/cols in range -> 8 unconditional stores
        int off = (ti * 16 + (hi ? 8 : 0)) * GASF_OUT + col;
#pragma unroll
        for (int v = 0; v < 8; ++v) {
          op[off] = acc[v];
          off += GASF_OUT;
        }
      } else {
        // ti == 4: only row 64 survives -> VGPR0, lo lane-half
        if (!hi) op[64 * GASF_OUT + col] = acc[0];
      }
    } else {
      // tj == 4: only col 64 survives -> lane m == 0
      if (m == 0) {
        if (ti < 4) {
          int off = (ti * 16 + (hi ? 8 : 0)) * GASF_OUT + 64;
#pragma unroll
          for (int v = 0; v < 8; ++v) {
            op[off] = acc[v];
            off += GASF_OUT;
          }
        } else {
          if (!hi) op[64 * GASF_OUT + 64] = acc[0];
        }
      }
    }
  }
}

extern "C" void kernel_launch(void* const* d_in, const int* in_sizes, int n_in,
                              void* d_out, int out_size, void* d_ws, size_t ws_size,
                              hipStream_t stream) {
  const float* x = (const float*)d_in[0];
  float* out = (float*)d_out;
  const int nc = in_sizes[0] / GASF_L;   // N*C = 2048 rows
  gasf_wmma_kernel<<<nc, 256, 0, stream>>>(x, out);
}